// GKDM_87428354278187
// MI455X (gfx1250) — compile-verified
//
#include <hip/hip_runtime.h>
#include <hip/hip_bf16.h>

#define C_ 256
#define H_ 384
#define W_ 384
#define HW_ (H_ * W_)       // 147456
#define K_ (C_ * 9)         // 2304

typedef __attribute__((ext_vector_type(2))) float v2f;
typedef __attribute__((ext_vector_type(4))) float v4f;
typedef __attribute__((ext_vector_type(8))) float v8f;

// ---------------------------------------------------------------------------
// Kernel 1: per-(input, channel) region sums.
// S[kh][kw] = sum over rows kh..kh+381, cols kw..kw+381 of x[c].
// One block per (channel, input), one thread per row.
// ---------------------------------------------------------------------------
__global__ __launch_bounds__(H_) void GKDM_region_sums(
    const float* __restrict__ in1, const float* __restrict__ in2,
    float* __restrict__ S /* [2][256][9] */) {
  const int c = blockIdx.x;       // channel
  const int which = blockIdx.y;   // 0 -> input1, 1 -> input2
  const float* src = (which == 0 ? in1 : in2) + (size_t)c * HW_;

  const int h = threadIdx.x;      // row 0..383
  const v4f* row = (const v4f*)(src + (size_t)h * W_);

  float sx = 0.f, sy = 0.f, sz = 0.f, sw = 0.f;
  v4f first = row[0];
  v4f last  = row[W_ / 4 - 1];
#pragma unroll 4
  for (int i = 0; i < W_ / 4; ++i) {
    v4f v = row[i];
    sx += v[0]; sy += v[1]; sz += v[2]; sw += v[3];
  }
  float s = (sx + sy) + (sz + sw);
  // column-shifted row sums: kw=0 drops cols 382,383; kw=1 drops 0,383; kw=2 drops 0,1
  float cs0 = s - last[2] - last[3];
  float cs1 = s - first[0] - last[3];
  float cs2 = s - first[0] - first[1];

  __shared__ float r0[H_], r1[H_], r2[H_];
  __shared__ float edge[4][3];    // rows 0,1,382,383
  r0[h] = cs0; r1[h] = cs1; r2[h] = cs2;
  if (h == 0)       { edge[0][0] = cs0; edge[0][1] = cs1; edge[0][2] = cs2; }
  if (h == 1)       { edge[1][0] = cs0; edge[1][1] = cs1; edge[1][2] = cs2; }
  if (h == H_ - 2)  { edge[2][0] = cs0; edge[2][1] = cs1; edge[2][2] = cs2; }
  if (h == H_ - 1)  { edge[3][0] = cs0; edge[3][1] = cs1; edge[3][2] = cs2; }
  __syncthreads();

  // 384 -> 128, then power-of-two tree (deterministic)
  if (h < 128) {
    r0[h] += r0[h + 128] + r0[h + 256];
    r1[h] += r1[h + 128] + r1[h + 256];
    r2[h] += r2[h + 128] + r2[h + 256];
  }
  __syncthreads();
  for (int off = 64; off >= 1; off >>= 1) {
    if (h < off) {
      r0[h] += r0[h + off];
      r1[h] += r1[h + off];
      r2[h] += r2[h + off];
    }
    __syncthreads();
  }

  if (h == 0) {
    float T[3] = { r0[0], r1[0], r2[0] };
    float* out = S + ((size_t)which * C_ + c) * 9;
#pragma unroll
    for (int kw = 0; kw < 3; ++kw) {
      out[0 * 3 + kw] = T[kw] - edge[2][kw] - edge[3][kw]; // rows 0..381
      out[1 * 3 + kw] = T[kw] - edge[0][kw] - edge[3][kw]; // rows 1..382
      out[2 * 3 + kw] = T[kw] - edge[0][kw] - edge[1][kw]; // rows 2..383
    }
  }
}

// ---------------------------------------------------------------------------
// Kernel 2: lin[o] = W[o,:] . S[:]  for S1 and S2 simultaneously via
// V_WMMA_F32_16X16X4_F32 (B columns 0/1 carry S1/S2), then per-channel
// branch decision from signs: g1 = adj1+adj3, g2 = adj2+adj3.
// One block, 8 waves, 2 output tiles (16 channels) per wave.
// B operand is staged in LDS with a zeroed pad region so every lane issues an
// unconditional ds_load_b64 (no per-iteration EXEC masking in the WMMA loop).
// ---------------------------------------------------------------------------
__global__ __launch_bounds__(256) void GKDM_gemv_decide(
    const float* __restrict__ Wt /* [256][2304] */,
    const float* __restrict__ S  /* [2][2304]   */,
    float* __restrict__ g        /* [2][256]    */) {
  __shared__ float buf[3 * K_];   // [0,K): S1, [K,2K): S2, [2K,3K): zeros
  __shared__ float lin[2][C_];

  const int tid = threadIdx.x;
  // Stage S1/S2 and the zero pad.
  for (int i = tid; i < K_; i += 256) buf[2 * K_ + i] = 0.f;
  for (int i = tid; i < 2 * K_; i += 256) buf[i] = S[i];
  __syncthreads();

  const int lane = tid & 31;
  const int wave = tid >> 5;
  const int n    = lane & 15;   // matrix row (A) / column (B) index
  const int half = lane >> 4;   // K half: 0 -> K=0,1 ; 1 -> K=2,3

  const int t0 = wave * 2, t1 = wave * 2 + 1;
  const float* rowA0 = Wt + (size_t)(t0 * 16 + n) * K_;
  const float* rowA1 = Wt + (size_t)(t1 * 16 + n) * K_;
  // Lanes n==0 read S1, n==1 read S2, all others read the zero region.
  const float* bBase = buf + (n == 0 ? 0 : (n == 1 ? K_ : 2 * K_));

  v8f acc0 = {};
  v8f acc1 = {};
  for (int k = 0; k < K_; k += 4) {
    const int ko = k + half * 2;
    // A layout (16x4 f32): lane M = lane&15 for both halves; VGPR0/1 = K pair
    v2f a0 = *(const v2f*)(rowA0 + ko);
    v2f a1 = *(const v2f*)(rowA1 + ko);
    // B layout (4x16 f32): lane N = lane&15; only columns 0 (S1) and 1 (S2) live
    v2f bv = *(const v2f*)(bBase + ko);
    acc0 = __builtin_amdgcn_wmma_f32_16x16x4_f32(false, a0, false, bv,
                                                 (short)0, acc0, false, false);
    acc1 = __builtin_amdgcn_wmma_f32_16x16x4_f32(false, a1, false, bv,
                                                 (short)0, acc1, false, false);
  }

  // D layout: VGPR v, lanes 0-15 -> M=v, lanes 16-31 -> M=v+8; N = lane&15.
  if (n < 2) {
#pragma unroll
    for (int v = 0; v < 8; ++v) {
      const int m = (half == 0) ? v : v + 8;
      lin[n][t0 * 16 + m] = acc0[v];
      lin[n][t1 * 16 + m] = acc1[v];
    }
  }
  __syncthreads();

  // Branch logic via signs (sigmoid & bias cancel):
  //   x1>=x2 <=> l1>=l2 ; x1>=x3 <=> l2<=0 ; x2>=x3 <=> l1<=0
  const int c = tid;
  const float l1 = lin[0][c];
  const float l2 = lin[1][c];
  const bool c1 = (l1 >= l2);
  const bool c2 = (l1 <= l2);
  const float adj1 = (c1 && l2 <= 0.f) ? 1.f : 0.f;
  const float adj2 = (c2 && l1 <= 0.f) ? 1.f : 0.f;
  const float adj3 = ((c1 && l2 > 0.f) || (c2 && l1 > 0.f)) ? 1.f : 0.f;
  g[c]      = adj1 + adj3;
  g[C_ + c] = adj2 + adj3;
}

// ---------------------------------------------------------------------------
// Kernel 3: out = g1[c]*in1 + g2[c]*in2 (bit-identical to the reference blend).
// Output stores are nontemporal so the 151 MB write stream doesn't evict the
// in1/in2 lines (302 MB vs 192 MB L2) warmed by kernel 1.
// ---------------------------------------------------------------------------
__global__ __launch_bounds__(256) void GKDM_combine(
    const float* __restrict__ in1, const float* __restrict__ in2,
    const float* __restrict__ g, float* __restrict__ out) {
  const int c = blockIdx.y;
  const float g1 = g[c];
  const float g2 = g[C_ + c];
  const size_t base = (size_t)c * HW_ + (size_t)blockIdx.x * 1024 + threadIdx.x * 4;
  const v4f a = *(const v4f*)(in1 + base);
  const v4f b = *(const v4f*)(in2 + base);
  v4f o;
  o[0] = g1 * a[0] + g2 * b[0];
  o[1] = g1 * a[1] + g2 * b[1];
  o[2] = g1 * a[2] + g2 * b[2];
  o[3] = g1 * a[3] + g2 * b[3];
  __builtin_nontemporal_store(o, (v4f*)(out + base));
}

extern "C" void kernel_launch(void* const* d_in, const int* in_sizes, int n_in,
                              void* d_out, int out_size, void* d_ws, size_t ws_size,
                              hipStream_t stream) {
  const float* in1 = (const float*)d_in[0];
  const float* in2 = (const float*)d_in[1];
  const float* Wt  = (const float*)d_in[2];   // [256,256,3,3]
  // d_in[3] = bias: not needed (cancels in all comparisons)
  float* out = (float*)d_out;

  float* S = (float*)d_ws;                    // [2][256][9] = 18432 B
  float* g = S + 2 * K_;                      // [2][256]    =  2048 B

  GKDM_region_sums<<<dim3(C_, 2), H_, 0, stream>>>(in1, in2, S);
  GKDM_gemv_decide<<<1, 256, 0, stream>>>(Wt, S, g);
  GKDM_combine<<<dim3(HW_ / 1024, C_), 256, 0, stream>>>(in1, in2, g, out);
}